// MuellerMatrixModel_72430328480040
// MI455X (gfx1250) — compile-verified
//
#include <hip/hip_runtime.h>
#include <cmath>

#define HDIM 512
#define WDIM 512
#define NB 2
#define PLANE (HDIM * WDIM)
#define NCH_OUT 21

typedef float v2f __attribute__((ext_vector_type(2)));
typedef float v8f __attribute__((ext_vector_type(8)));

__device__ __forceinline__ int reflect_idx(int i) {
  i = (i < 0) ? -i : i;
  i = (i > (HDIM - 1)) ? (2 * (HDIM - 1) - i) : i;
  return i;
}

__device__ __forceinline__ float clampf(float v, float lo, float hi) {
  return fminf(fmaxf(v, lo), hi);
}

// 4x4 inverse (adjugate). Feeding row-major data to this col-major-convention
// routine returns the row-major inverse (inv(M^T)^T == inv(M)).
__device__ __forceinline__ void inv4(const float m[16], float inv[16]) {
  float i0  =  m[5]*m[10]*m[15] - m[5]*m[11]*m[14] - m[9]*m[6]*m[15] + m[9]*m[7]*m[14] + m[13]*m[6]*m[11] - m[13]*m[7]*m[10];
  float i4  = -m[4]*m[10]*m[15] + m[4]*m[11]*m[14] + m[8]*m[6]*m[15] - m[8]*m[7]*m[14] - m[12]*m[6]*m[11] + m[12]*m[7]*m[10];
  float i8  =  m[4]*m[9]*m[15]  - m[4]*m[11]*m[13] - m[8]*m[5]*m[15] + m[8]*m[7]*m[13] + m[12]*m[5]*m[11] - m[12]*m[7]*m[9];
  float i12 = -m[4]*m[9]*m[14]  + m[4]*m[10]*m[13] + m[8]*m[5]*m[14] - m[8]*m[6]*m[13] - m[12]*m[5]*m[10] + m[12]*m[6]*m[9];
  float i1  = -m[1]*m[10]*m[15] + m[1]*m[11]*m[14] + m[9]*m[2]*m[15] - m[9]*m[3]*m[14] - m[13]*m[2]*m[11] + m[13]*m[3]*m[10];
  float i5  =  m[0]*m[10]*m[15] - m[0]*m[11]*m[14] - m[8]*m[2]*m[15] + m[8]*m[3]*m[14] + m[12]*m[2]*m[11] - m[12]*m[3]*m[10];
  float i9  = -m[0]*m[9]*m[15]  + m[0]*m[11]*m[13] + m[8]*m[1]*m[15] - m[8]*m[3]*m[13] - m[12]*m[1]*m[11] + m[12]*m[3]*m[9];
  float i13 =  m[0]*m[9]*m[14]  - m[0]*m[10]*m[13] - m[8]*m[1]*m[14] + m[8]*m[2]*m[13] - m[12]*m[2]*m[9]  + m[12]*m[1]*m[10];
  float i2  =  m[1]*m[6]*m[15]  - m[1]*m[7]*m[14]  - m[5]*m[2]*m[15] + m[5]*m[3]*m[14] + m[13]*m[2]*m[7]  - m[13]*m[3]*m[6];
  float i6  = -m[0]*m[6]*m[15]  + m[0]*m[7]*m[14]  + m[4]*m[2]*m[15] - m[4]*m[3]*m[14] - m[12]*m[2]*m[7]  + m[12]*m[3]*m[6];
  float i10 =  m[0]*m[5]*m[15]  - m[0]*m[7]*m[13]  - m[4]*m[1]*m[15] + m[4]*m[3]*m[13] + m[12]*m[1]*m[7]  - m[12]*m[3]*m[5];
  float i14 = -m[0]*m[5]*m[14]  + m[0]*m[6]*m[13]  + m[4]*m[1]*m[14] - m[4]*m[2]*m[13] - m[12]*m[1]*m[6]  + m[12]*m[2]*m[5];
  float i3  = -m[1]*m[6]*m[11]  + m[1]*m[7]*m[10]  + m[5]*m[2]*m[11] - m[5]*m[3]*m[10] - m[9]*m[2]*m[7]   + m[9]*m[3]*m[6];
  float i7  =  m[0]*m[6]*m[11]  - m[0]*m[7]*m[10]  - m[4]*m[2]*m[11] + m[4]*m[3]*m[10] + m[8]*m[2]*m[7]   - m[8]*m[3]*m[6];
  float i11 = -m[0]*m[5]*m[11]  + m[0]*m[7]*m[9]   + m[4]*m[1]*m[11] - m[4]*m[3]*m[9]  - m[8]*m[1]*m[7]   + m[8]*m[3]*m[5];
  float i15 =  m[0]*m[5]*m[10]  - m[0]*m[6]*m[9]   - m[4]*m[1]*m[10] + m[4]*m[2]*m[9]  + m[8]*m[1]*m[6]   - m[8]*m[2]*m[5];
  float det = m[0]*i0 + m[1]*i4 + m[2]*i8 + m[3]*i12;
  float r = 1.0f / det;
  inv[0]=i0*r;  inv[1]=i1*r;  inv[2]=i2*r;  inv[3]=i3*r;
  inv[4]=i4*r;  inv[5]=i5*r;  inv[6]=i6*r;  inv[7]=i7*r;
  inv[8]=i8*r;  inv[9]=i9*r;  inv[10]=i10*r; inv[11]=i11*r;
  inv[12]=i12*r; inv[13]=i13*r; inv[14]=i14*r; inv[15]=i15*r;
}

__device__ __forceinline__ void mat4mul(const float a[16], const float b[16], float c[16]) {
#pragma unroll
  for (int i = 0; i < 4; ++i)
#pragma unroll
    for (int j = 0; j < 4; ++j) {
      float s = 0.0f;
#pragma unroll
      for (int k = 0; k < 4; ++k) s += a[i*4+k] * b[k*4+j];
      c[i*4+j] = s;
    }
}

__device__ __forceinline__ float det3(const float a[9]) {
  return a[0]*(a[4]*a[8]-a[5]*a[7]) - a[1]*(a[3]*a[8]-a[5]*a[6]) + a[2]*(a[3]*a[7]-a[4]*a[6]);
}

__device__ __forceinline__ void inv3(const float a[9], float o[9]) {
  float c0 = a[4]*a[8]-a[5]*a[7];
  float c1 = a[5]*a[6]-a[3]*a[8];
  float c2 = a[3]*a[7]-a[4]*a[6];
  float det = a[0]*c0 + a[1]*c1 + a[2]*c2;
  float r = 1.0f/det;
  o[0]=c0*r;                     o[1]=(a[2]*a[7]-a[1]*a[8])*r; o[2]=(a[1]*a[5]-a[2]*a[4])*r;
  o[3]=c1*r;                     o[4]=(a[0]*a[8]-a[2]*a[6])*r; o[5]=(a[2]*a[3]-a[0]*a[5])*r;
  o[6]=c2*r;                     o[7]=(a[1]*a[6]-a[0]*a[7])*r; o[8]=(a[0]*a[4]-a[1]*a[3])*r;
}

__device__ __forceinline__ void mat3mul(const float a[9], const float b[9], float c[9]) {
#pragma unroll
  for (int i = 0; i < 3; ++i)
#pragma unroll
    for (int j = 0; j < 3; ++j) {
      float s = 0.0f;
#pragma unroll
      for (int k = 0; k < 3; ++k) s += a[i*3+k]*b[k*3+j];
      c[i*3+j] = s;
    }
}

// ---------------------------------------------------------------------------
// Kernel 1: fused per-pixel Mueller pipeline. One lane per pixel (pure VALU:
// lane-local 4x4 / 3x3 algebra beats any WMMA packing for per-pixel tiny mats).
// Writes output channels 0..19; channel 20 (std) comes from the filter kernels.
// ---------------------------------------------------------------------------
__global__ void mueller_pixel_kernel(const float* __restrict__ x, float* __restrict__ out) {
  int idx = blockIdx.x * blockDim.x + threadIdx.x;     // 0 .. NB*PLANE-1
  int xc = idx & (WDIM - 1);
  int y  = (idx >> 9) & (HDIM - 1);
  int bb = idx >> 18;
  const size_t plane = (size_t)PLANE;
  const float* base = x + (size_t)bb * 48 * plane + (size_t)y * WDIM + xc;

  float Im[16], Am[16], Wm[16];
#pragma unroll
  for (int c = 0; c < 16; ++c) {
    Im[c] = base[(size_t)c * plane];
    Am[c] = base[(size_t)(16 + c) * plane];
    Wm[c] = base[(size_t)(32 + c) * plane];
  }

  float intensity = 0.0f;
#pragma unroll
  for (int c = 0; c < 16; ++c) intensity += Im[c];
  intensity *= (1.0f / 16.0f);

  // M = inv(A) @ Imeas @ inv(W)
  float iA[16], iW[16], T[16], M[16];
  inv4(Am, iA);
  inv4(Wm, iW);
  mat4mul(iA, Im, T);
  mat4mul(T, iW, M);

  // ---- char-poly physicality mask via Pauli coherency matrix ----
  float Hr[4][4] = {{0}}, Hi[4][4] = {{0}};
  {
    // Pauli order of reference: I, Z, X, Y
    constexpr float SRl[4][2][2] = {
      {{1.f,0.f},{0.f,1.f}},
      {{1.f,0.f},{0.f,-1.f}},
      {{0.f,1.f},{1.f,0.f}},
      {{0.f,0.f},{0.f,0.f}}};
    constexpr float SIl[4][2][2] = {
      {{0.f,0.f},{0.f,0.f}},
      {{0.f,0.f},{0.f,0.f}},
      {{0.f,0.f},{0.f,0.f}},
      {{0.f,-1.f},{1.f,0.f}}};
#pragma unroll
    for (int i = 0; i < 4; ++i)
#pragma unroll
      for (int j = 0; j < 4; ++j) {
        float mv = 0.25f * M[i*4+j];
#pragma unroll
        for (int a1 = 0; a1 < 2; ++a1)
#pragma unroll
          for (int b1 = 0; b1 < 2; ++b1)
#pragma unroll
            for (int a2 = 0; a2 < 2; ++a2)
#pragma unroll
              for (int b2 = 0; b2 < 2; ++b2) {
                float sr = SRl[i][a1][b1], si = SIl[i][a1][b1];
                float cr = SRl[j][a2][b2], ci = -SIl[j][a2][b2];   // conj
                float pr  = sr*cr - si*ci;
                float pim = sr*ci + si*cr;
                Hr[2*a1+a2][2*b1+b2] += mv * pr;
                Hi[2*a1+a2][2*b1+b2] += mv * pim;
              }
      }
  }
  float H2r[4][4], H2i[4][4];
#pragma unroll
  for (int i = 0; i < 4; ++i)
#pragma unroll
    for (int j = 0; j < 4; ++j) {
      float ar = 0.0f, ai = 0.0f;
#pragma unroll
      for (int k = 0; k < 4; ++k) {
        ar += Hr[i][k]*Hr[k][j] - Hi[i][k]*Hi[k][j];
        ai += Hr[i][k]*Hi[k][j] + Hi[i][k]*Hr[k][j];
      }
      H2r[i][j] = ar; H2i[i][j] = ai;
    }
  float p1 = Hr[0][0]+Hr[1][1]+Hr[2][2]+Hr[3][3];
  float p2 = H2r[0][0]+H2r[1][1]+H2r[2][2]+H2r[3][3];
  float p3 = 0.0f, p4 = 0.0f;
#pragma unroll
  for (int i = 0; i < 4; ++i)
#pragma unroll
    for (int j = 0; j < 4; ++j) {
      p3 += H2r[i][j]*Hr[i][j] + H2i[i][j]*Hi[i][j];
      p4 += H2r[i][j]*H2r[i][j] + H2i[i][j]*H2i[i][j];
    }
  float c1 = p1;
  float c2 = (c1*p1 - p2) * 0.5f;
  float c3 = (c2*p1 - c1*p2 + p3) * (1.0f/3.0f);
  float c4 = (c3*p1 - c2*p2 + c1*p3 - p4) * 0.25f;
  const float tol = -1e-5f;
  bool v = (c1 > tol) && (c2 > tol) && (c3 > tol) && (c4 > tol);

  // ---- Lu-Chipman polar decomposition ----
  float inv00 = 1.0f / (M[0] + 1e-6f);
  float Mn[16];
#pragma unroll
  for (int k = 0; k < 16; ++k) Mn[k] = M[k] * inv00;
  float D0 = Mn[1], D1 = Mn[2], D2 = Mn[3];
  float d   = sqrtf(D0*D0 + D1*D1 + D2*D2 + 1e-12f);
  float dcl = clampf(d, 0.0f, 1.0f - 1e-4f);
  float sq  = sqrtf(1.0f - dcl*dcl);
  float invd = 1.0f / (d + 1e-9f);
  float Dn0 = D0*invd, Dn1 = D1*invd, Dn2 = D2*invd;
  float os = 1.0f - sq;
  float MDm[16] = {
    1.0f, D0, D1, D2,
    D0, sq + os*Dn0*Dn0, os*Dn0*Dn1,       os*Dn0*Dn2,
    D1, os*Dn1*Dn0,      sq + os*Dn1*Dn1,  os*Dn1*Dn2,
    D2, os*Dn2*Dn0,      os*Dn2*Dn1,       sq + os*Dn2*Dn2 };
  float iMD[16], Mp[16];
  inv4(MDm, iMD);
  mat4mul(Mn, iMD, Mp);
  float mp[9] = { Mp[5], Mp[6], Mp[7], Mp[9], Mp[10], Mp[11], Mp[13], Mp[14], Mp[15] };
  float G[9];
#pragma unroll
  for (int i = 0; i < 3; ++i)
#pragma unroll
    for (int j = 0; j < 3; ++j) {
      float s = 0.0f;
#pragma unroll
      for (int k = 0; k < 3; ++k) s += mp[i*3+k]*mp[j*3+k];
      G[i*3+j] = s;
    }

  // symmetric 3x3 eigenvalues (trigonometric)
  float q = (G[0]+G[4]+G[8]) * (1.0f/3.0f);
  float Gq[9]; 
#pragma unroll
  for (int k = 0; k < 9; ++k) Gq[k] = G[k];
  Gq[0] -= q; Gq[4] -= q; Gq[8] -= q;
  float ssum = 0.0f;
#pragma unroll
  for (int k = 0; k < 9; ++k) ssum += Gq[k]*Gq[k];
  float pp  = sqrtf(ssum * (1.0f/6.0f) + 1e-20f);
  float rr  = clampf(det3(Gq) / (2.0f*pp*pp*pp), -1.0f + 1e-6f, 1.0f - 1e-6f);
  float phi = acosf(rr) * (1.0f/3.0f);
  float l1 = q + 2.0f*pp*cosf(phi);
  float l3 = q + 2.0f*pp*cosf(phi + 2.0943951023931953f);
  float l2 = 3.0f*q - l1 - l3;
  float s1 = sqrtf(fmaxf(l1, 1e-12f));
  float s2 = sqrtf(fmaxf(l2, 1e-12f));
  float s3 = sqrtf(fmaxf(l3, 1e-12f));
  float e1 = s1+s2+s3, e2 = s1*s2 + s2*s3 + s3*s1, e3 = s1*s2*s3;
  float sgn = (det3(mp) >= 0.0f) ? 1.0f : -1.0f;

  float Km[9], Nm[9], iK[9], mDl[9];
#pragma unroll
  for (int k = 0; k < 9; ++k) { Km[k] = G[k]; Nm[k] = e1*G[k]; }
  float ed = e2 + 1e-9f;
  Km[0] += ed; Km[4] += ed; Km[8] += ed;
  Nm[0] += e3; Nm[4] += e3; Nm[8] += e3;
  inv3(Km, iK);
  mat3mul(iK, Nm, mDl);
#pragma unroll
  for (int k = 0; k < 9; ++k) mDl[k] *= sgn;

  float iDl[9], mR[9];
  inv3(mDl, iDl);
  mat3mul(iDl, mp, mR);

  float targ = sqrtf((mR[0]+mR[4])*(mR[0]+mR[4]) + (mR[3]-mR[1])*(mR[3]-mR[1]) + 1e-12f) - 1.0f;
  targ = clampf(targ, -1.0f + 1e-6f, 1.0f - 1e-6f);
  float linr = 57.29577951308232f * acosf(targ);
  float a1v = mR[5] - mR[7];
  float a2v = mR[6] - mR[2];
  bool safe = (a1v*a1v + a2v*a2v) > 1e-12f;
  float azi = 0.5f * 57.29577951308232f * atan2f(safe ? a2v : 0.0f, safe ? a1v : 1.0f);
  azi = fmodf(azi, 180.0f);
  if (azi < 0.0f) azi += 180.0f;
  float totp = 1.0f - (fabsf(mDl[0]) + fabsf(mDl[4]) + fabsf(mDl[8])) * (1.0f/3.0f);
  if (!v) { linr = 0.0f; totp = 0.0f; azi = 0.0f; }

  float* ob = out + (size_t)bb * NCH_OUT * plane + (size_t)y * WDIM + xc;
  ob[0] = intensity;
#pragma unroll
  for (int k = 0; k < 16; ++k) ob[(size_t)(1 + k) * plane] = M[k];
  ob[17 * plane] = linr;
  ob[18 * plane] = totp;
  ob[19 * plane] = azi;
}

// ---------------------------------------------------------------------------
// Kernel 2: horizontal 8-tap reflect box sum of cos/sin(2*radians(azi)).
// Reads masked azi from output channel 19 (avoids extra ws buffers).
// ---------------------------------------------------------------------------
__global__ void box_h_kernel(const float* __restrict__ out,
                             float* __restrict__ hC, float* __restrict__ hS) {
  int idx = blockIdx.x * blockDim.x + threadIdx.x;
  int xc = idx & (WDIM - 1);
  int y  = (idx >> 9) & (HDIM - 1);
  int bb = idx >> 18;
  const float* azip = out + ((size_t)bb * NCH_OUT + 19) * PLANE + (size_t)y * WDIM;
  float sc = 0.0f, sn = 0.0f;
#pragma unroll
  for (int dx = 0; dx < 8; ++dx) {
    int xi = reflect_idx(xc - 4 + dx);
    float ang = azip[xi] * 0.034906585039886591f;  // 2 * pi/180
    sc += cosf(ang);
    sn += sinf(ang);
  }
  size_t o = (size_t)bb * PLANE + (size_t)y * WDIM + xc;
  hC[o] = sc;
  hS[o] = sn;
}

__device__ __forceinline__ float band8(int k, int m) {
  int dd = k - m;
  return (dd >= 0 && dd <= 7) ? 1.0f : 0.0f;
}

// ---------------------------------------------------------------------------
// Kernel 3: vertical 8-tap reflect box sum expressed as a banded matmul on
// the matrix core: D[16,16] = Band[16,24] x Rows[24,16] via six chained
// V_WMMA_F32_16X16X4_F32 per channel (f32 accumulation => exact sums), then
// circular-std finalize. One wave (32 lanes) per 16x16 tile, EXEC all ones.
// Operand layouts per CDNA5 ISA 7.12.2:
//   A 16x4:  lanes 0-15 -> M=lane, K={0,1}; lanes 16-31 -> M=lane-16, K={2,3}
//   B 4x16:  lanes 0-15 -> K={0,1}, N=lane; lanes 16-31 -> K={2,3}, N=lane-16
//   C/D:     VGPR v: lanes 0-15 -> M=v, N=lane; lanes 16-31 -> M=v+8
// ---------------------------------------------------------------------------
__global__ void box_v_std_kernel(const float* __restrict__ hC,
                                 const float* __restrict__ hS,
                                 float* __restrict__ out) {
  const int lane = threadIdx.x;              // 0..31
  const int x0 = blockIdx.x * 16;
  const int y0 = blockIdx.y * 16;
  const int bz = blockIdx.z;
  const int m  = lane & 15;
  const int hi = (lane >= 16) ? 1 : 0;
  const int col = x0 + m;
  const size_t cbase = (size_t)bz * PLANE;

  v8f accC = {};
  v8f accS = {};
#pragma unroll
  for (int c = 0; c < 6; ++c) {
    const int k0 = 4 * c + (hi ? 2 : 0);     // K index held by this half-wave
    const int k1 = k0 + 1;
    v2f a;
    a.x = band8(k0, m);                      // Band[m, k] = 1 iff 0 <= k-m <= 7
    a.y = band8(k1, m);
    const int r0 = reflect_idx(y0 - 4 + k0);
    const int r1 = reflect_idx(y0 - 4 + k1);
    v2f bc, bs;
    bc.x = hC[cbase + (size_t)r0 * WDIM + col];
    bc.y = hC[cbase + (size_t)r1 * WDIM + col];
    bs.x = hS[cbase + (size_t)r0 * WDIM + col];
    bs.y = hS[cbase + (size_t)r1 * WDIM + col];
    accC = __builtin_amdgcn_wmma_f32_16x16x4_f32(false, a, false, bc, (short)0, accC, false, false);
    accS = __builtin_amdgcn_wmma_f32_16x16x4_f32(false, a, false, bs, (short)0, accS, false, false);
  }

  float* ob = out + ((size_t)bz * NCH_OUT + 20) * PLANE;
#pragma unroll
  for (int vv = 0; vv < 8; ++vv) {
    const int y = y0 + vv + (hi ? 8 : 0);
    float Cm = accC[vv] * (1.0f / 64.0f);
    float Sm = accS[vv] * (1.0f / 64.0f);
    float R = sqrtf(Cm*Cm + Sm*Sm + 1e-12f);
    R = clampf(R, 1e-6f, 1.0f - 1e-7f);
    float stdv = 57.29577951308232f * 0.5f * sqrtf(-2.0f * logf(R));
    ob[(size_t)y * WDIM + col] = stdv;
  }
}

extern "C" void kernel_launch(void* const* d_in, const int* in_sizes, int n_in,
                              void* d_out, int out_size, void* d_ws, size_t ws_size,
                              hipStream_t stream) {
  (void)in_sizes; (void)n_in; (void)out_size; (void)ws_size;
  const float* x = (const float*)d_in[0];
  float* out = (float*)d_out;
  float* hC = (float*)d_ws;                       // NB*PLANE floats
  float* hS = hC + (size_t)NB * PLANE;            // NB*PLANE floats (ws >= 4 MB)

  const int npix = NB * PLANE;                    // 524288
  mueller_pixel_kernel<<<npix / 256, 256, 0, stream>>>(x, out);
  box_h_kernel<<<npix / 256, 256, 0, stream>>>(out, hC, hS);
  dim3 gridC(WDIM / 16, HDIM / 16, NB);
  box_v_std_kernel<<<gridC, 32, 0, stream>>>(hC, hS, out);
}